// RNNLM_13657996001435
// MI455X (gfx1250) — compile-verified
//
#include <hip/hip_runtime.h>
#include <hip/hip_bf16.h>

// CDNA5 wave32 WMMA types
typedef __attribute__((ext_vector_type(2))) float v2f;
typedef __attribute__((ext_vector_type(8))) float v8f;

#define D_ 512
#define FOURD 2048
#define T_ 1024
#define B_ 4
#define NWG_SCAN 16

// ---------------------------------------------------------------------------
// fp32 WMMA helper: D = A(16x4) * B(4x16) + C  via V_WMMA_F32_16X16X4_F32
// A frag (per lane): m = lane&15, a.x = A[m][k0+2*half], a.y = A[m][k0+2*half+1]
// B frag (per lane): n = lane&15, b.x = B[k0+2*half][n], b.y = B[k0+2*half+1][n]
// C/D: vgpr j holds (m = j + 8*half, n = lane&15)
// ---------------------------------------------------------------------------
__device__ __forceinline__ v8f wmma4(v2f a, v2f b, v8f c) {
  return __builtin_amdgcn_wmma_f32_16x16x4_f32(false, a, false, b, (short)0, c,
                                               false, false);
}

// ---------------------------------------------------------------------------
// init: reset inter-block sync state (workspace is poisoned by harness)
// ---------------------------------------------------------------------------
__global__ void init_sync(unsigned* __restrict__ sync) {
  if (threadIdx.x < 2) sync[threadIdx.x] = 0u;
}

// ---------------------------------------------------------------------------
// Embedding gather with ShiftRight: x[b,t,:] = embed[t==0 ? 0 : tokens[b,t-1]]
// ---------------------------------------------------------------------------
__global__ __launch_bounds__(128) void embed_gather(
    const int* __restrict__ tokens, const float* __restrict__ embed,
    float* __restrict__ x) {
  const int bt = blockIdx.x;          // b*T + t
  const int t = bt & (T_ - 1);
  const int tok = (t == 0) ? 0 : tokens[bt - 1];
  const float4* src = (const float4*)(embed + (long)tok * D_);
  float4* dst = (float4*)(x + (long)bt * D_);
  dst[threadIdx.x] = src[threadIdx.x];
}

// ---------------------------------------------------------------------------
// C[M,N] = A[M,K] @ B[K,N] + bias[N]   (row-major, fp32, WMMA 16x16x4)
// Workgroup: 256 threads = 8 waves; tile 256 rows x 64 cols
// wave w: rows [by*256 + 32w, +32) (2 m-tiles), cols [bx*64, +64) (4 n-tiles)
// A/B stream through L2 (all operands fit in 192MB L2).
// ---------------------------------------------------------------------------
__global__ __launch_bounds__(256) void wmma_gemm_bias(
    const float* __restrict__ A, const float* __restrict__ B,
    const float* __restrict__ bias, float* __restrict__ C,
    int N, int K) {
  const int lane = threadIdx.x & 31;
  const int wv = threadIdx.x >> 5;
  const int half = lane >> 4;
  const int l16 = lane & 15;
  const int rowbase = blockIdx.y * 256 + wv * 32;
  const int colbase = blockIdx.x * 64;

  const v8f vzero = {0.f, 0.f, 0.f, 0.f, 0.f, 0.f, 0.f, 0.f};
  v8f acc[2][4];
#pragma unroll
  for (int i = 0; i < 2; ++i)
#pragma unroll
    for (int j = 0; j < 4; ++j) acc[i][j] = vzero;

  const float* Ar0 = A + (long)(rowbase + l16) * K;
  const float* Ar1 = Ar0 + 16L * K;

#pragma unroll 2
  for (int k4 = 0; k4 < K; k4 += 4) {
    const int ka = k4 + 2 * half;
    v2f a0 = *(const v2f*)(Ar0 + ka);
    v2f a1 = *(const v2f*)(Ar1 + ka);
    const float* Bp = B + (long)ka * N + colbase + l16;
#pragma unroll
    for (int nt = 0; nt < 4; ++nt) {
      v2f b;
      b.x = Bp[nt * 16];
      b.y = Bp[nt * 16 + N];
      acc[0][nt] = wmma4(a0, b, acc[0][nt]);
      acc[1][nt] = wmma4(a1, b, acc[1][nt]);
    }
  }

#pragma unroll
  for (int mt = 0; mt < 2; ++mt)
#pragma unroll
    for (int nt = 0; nt < 4; ++nt) {
      const int n = colbase + nt * 16 + l16;
      const float bv = bias[n];
#pragma unroll
      for (int j = 0; j < 8; ++j) {
        const int row = rowbase + mt * 16 + j + 8 * half;
        C[(long)row * N + n] = acc[mt][nt][j] + bv;
      }
    }
}

// ---------------------------------------------------------------------------
// Device-scope barrier across the 16 persistent scan workgroups
// ---------------------------------------------------------------------------
__device__ __forceinline__ void grid_barrier(unsigned* sync, unsigned nwg) {
  __syncthreads();
  if (threadIdx.x == 0) {
    const unsigned g = __atomic_load_n(sync + 1, __ATOMIC_ACQUIRE);
    __threadfence();
    if (atomicAdd(sync, 1u) == nwg - 1u) {
      __atomic_store_n(sync, 0u, __ATOMIC_RELAXED);
      __threadfence();
      __atomic_store_n(sync + 1, g + 1u, __ATOMIC_RELEASE);
    } else {
      while (__atomic_load_n(sync + 1, __ATOMIC_ACQUIRE) == g)
        __builtin_amdgcn_s_sleep(1);
    }
    __threadfence();
  }
  __syncthreads();
}

// ---------------------------------------------------------------------------
// Persistent LSTM scan. Grid = 16 WGs x 256 threads (8 waves).
// WG j owns h dims [32j, 32j+32). Its 128 Wh columns (i/f/g/o slices of 32)
// are preloaded into LDS (512x128 fp32 = 256KB, CDNA5 has 320KB/WGP).
// Per step: load h (4x512, padded to 16 rows of zeros) into LDS, each wave
// does a 16x16 gate tile via chained v_wmma_f32_16x16x4_f32 reading A and B
// from LDS -- split across TWO accumulators to halve the RAW dependency
// chain (the scan is latency-bound) -- then activations + c/h update, then
// one device-scope barrier.
// LDS gate-column index cc maps to global column (cc>>5)*512 + 32j + (cc&31).
// ---------------------------------------------------------------------------
__global__ __launch_bounds__(256) void lstm_scan(
    const float* __restrict__ xg,   // [B*T, 4D] (bias already folded in)
    const float* __restrict__ Wh,   // [D, 4D]
    float* __restrict__ out,        // [B*T, D]
    float* __restrict__ h_buf,      // [B, D] exchange buffer
    unsigned* __restrict__ sync) {
  extern __shared__ float smem[];
  float* whl  = smem;                  // [512][128]
  float* ldsh = whl + 512 * 128;       // [16][512]  (rows 4..15 stay zero)
  float* gl   = ldsh + 16 * 512;       // [4][128]   gate pre-activations
  float* cst  = gl + 4 * 128;          // [4][32]    cell state

  const int tid = threadIdx.x;
  const int wg = blockIdx.x;           // 0..15
  const int lane = tid & 31;
  const int wv = tid >> 5;
  const int half = lane >> 4;
  const int l16 = lane & 15;

  // Preload this WG's Wh slice into LDS (once; reused for 1024 steps).
  for (int e = tid; e < 512 * 128; e += 256) {
    const int k = e >> 7;
    const int cc = e & 127;
    const int col = ((cc >> 5) << 9) + wg * 32 + (cc & 31);
    whl[e] = Wh[(long)k * FOURD + col];
  }
  // Zero padded h tile (batch rows 4..15 remain zero forever; rows 0..3
  // are zero for t==0 and overwritten for t>0).
  for (int e = tid; e < 16 * 512; e += 256) ldsh[e] = 0.f;
  if (tid < 128) cst[tid] = 0.f;
  __syncthreads();

  const int cc = wv * 16 + l16;        // this wave's gate column (LDS space)
  const float* ar = ldsh + l16 * 512;  // A row m = lane&15 (batch, padded)
  const v8f vzero = {0.f, 0.f, 0.f, 0.f, 0.f, 0.f, 0.f, 0.f};

  for (int t = 0; t < T_; ++t) {
    if (t > 0) {
      for (int e = tid; e < B_ * D_; e += 256) ldsh[e] = h_buf[e];
    }
    __syncthreads();

    // gates[0:16, cc] = h_pad @ Wh_slice  (K = 512; two independent WMMA
    // chains of 64 for 2x ILP in the latency-critical recurrence)
    v8f acc0 = vzero;
    v8f acc1 = vzero;
#pragma unroll 2
    for (int k4 = 0; k4 < D_; k4 += 8) {
      {
        const int ka = k4 + 2 * half;
        v2f a = *(const v2f*)(ar + ka);
        v2f b;
        b.x = whl[ka * 128 + cc];
        b.y = whl[(ka + 1) * 128 + cc];
        acc0 = wmma4(a, b, acc0);
      }
      {
        const int kb = k4 + 4 + 2 * half;
        v2f a = *(const v2f*)(ar + kb);
        v2f b;
        b.x = whl[kb * 128 + cc];
        b.y = whl[(kb + 1) * 128 + cc];
        acc1 = wmma4(a, b, acc1);
      }
    }
    const v8f acc = acc0 + acc1;

    // valid batch rows m=0..3 live in vgprs 0..3 of lanes 0..15
    if (lane < 16) {
#pragma unroll
      for (int m = 0; m < 4; ++m) gl[m * 128 + cc] = acc[m];
    }
    __syncthreads();

    // Elementwise LSTM cell update for this WG's 4 batches x 32 dims.
    if (tid < 128) {
      const int bb = tid >> 5;
      const int dl = tid & 31;
      const int d = wg * 32 + dl;
      const float* xgp = xg + (long)(bb * T_ + t) * FOURD;
      const float gi = gl[bb * 128 + 0 + dl]  + xgp[0 * D_ + d];
      const float gf = gl[bb * 128 + 32 + dl] + xgp[1 * D_ + d];
      const float gg = gl[bb * 128 + 64 + dl] + xgp[2 * D_ + d];
      const float go = gl[bb * 128 + 96 + dl] + xgp[3 * D_ + d];
      const float si = 1.f / (1.f + __expf(-gi));
      const float sf = 1.f / (1.f + __expf(-gf));
      const float so = 1.f / (1.f + __expf(-go));
      const float cn = sf * cst[tid] + si * tanhf(gg);
      cst[tid] = cn;
      const float hn = so * tanhf(cn);
      h_buf[bb * D_ + d] = hn;
      out[(long)(bb * T_ + t) * D_ + d] = hn;
    }
    __threadfence();
    grid_barrier(sync, NWG_SCAN);
  }
}

// ---------------------------------------------------------------------------
// Host launch
// ---------------------------------------------------------------------------
extern "C" void kernel_launch(void* const* d_in, const int* in_sizes, int n_in,
                              void* d_out, int out_size, void* d_ws,
                              size_t ws_size, hipStream_t stream) {
  const int*   tokens = (const int*)d_in[0];
  const float* embed  = (const float*)d_in[1];
  const float* Wx0    = (const float*)d_in[2];
  const float* Wh0    = (const float*)d_in[3];
  const float* b0     = (const float*)d_in[4];
  const float* Wx1    = (const float*)d_in[5];
  const float* Wh1    = (const float*)d_in[6];
  const float* b1     = (const float*)d_in[7];
  const float* Wout   = (const float*)d_in[8];
  const float* bout   = (const float*)d_in[9];
  float* logits = (float*)d_out;

  // Workspace layout (bytes)
  char* ws = (char*)d_ws;
  float* x_in  = (float*)(ws);                         //  4096*512  =  8 MiB
  float* xg    = (float*)(ws + (8u << 20));            //  4096*2048 = 32 MiB
  float* x_mid = (float*)(ws + (40u << 20));           //  4096*512  =  8 MiB
  float* h_buf = (float*)(ws + (48u << 20));           //  4*512
  unsigned* sync = (unsigned*)(ws + (48u << 20) + 8192);

  const size_t scan_lds =
      (512 * 128 + 16 * 512 + 4 * 128 + 4 * 32) * sizeof(float);  // ~291 KiB

  init_sync<<<1, 32, 0, stream>>>(sync);
  embed_gather<<<B_ * T_, 128, 0, stream>>>(tokens, embed, x_in);

  dim3 g_xg(FOURD / 64, (B_ * T_) / 256);
  // Layer 0
  wmma_gemm_bias<<<g_xg, 256, 0, stream>>>(x_in, Wx0, b0, xg, FOURD, D_);
  lstm_scan<<<NWG_SCAN, 256, scan_lds, stream>>>(xg, Wh0, x_mid, h_buf, sync);
  // Layer 1
  wmma_gemm_bias<<<g_xg, 256, 0, stream>>>(x_mid, Wx1, b1, xg, FOURD, D_);
  lstm_scan<<<NWG_SCAN, 256, scan_lds, stream>>>(xg, Wh1, x_in, h_buf, sync);
  // Output projection (bandwidth-bound: 524 MB of logits)
  dim3 g_out(32000 / 64, (B_ * T_) / 256);
  wmma_gemm_bias<<<g_out, 256, 0, stream>>>(x_in, Wout, bout, logits, 32000, D_);
}